// QuantumPolicySimple_56135222558893
// MI455X (gfx1250) — compile-verified
//
#include <hip/hip_runtime.h>

// ============================================================================
// QuantumPolicySimple on MI455X (gfx1250)
//
// Math reduction (done once per launch on 1 wave, then a pure streaming pass):
//   action(b) = h(b)^T * S' * g(b),
//     h = (1, cos(pi*tanh f0), sin(pi*tanh f0)),
//     g = (1, cos(pi*tanh f1), sin(pi*tanh f1)),
//   where S' (3x3) is derived from the weight-only 4x4 unitary U via
//   Q = M_re^T D M_re + M_im^T D M_im  (computed with v_wmma_f32_16x16x4_f32)
//   and a double-angle bilinear reduction.
//
// Roofline: 4.19M * (16B read + 4B write) = 84 MB -> ~3.6 us @ 23.3 TB/s.
// Streaming kernel per element: 1x b128 load, 2 tanh + 4 sin/cos trans ops,
// 8 FMAs, 1x b32 store. Memory-bound; keep RT caching (84 MB < 192 MB L2).
// ============================================================================

#define N_LAYERS 4

typedef __attribute__((ext_vector_type(2))) float v2f;
typedef __attribute__((ext_vector_type(8))) float v8f;

struct C2 { float re, im; };
__device__ __forceinline__ C2 cmul(C2 a, C2 b) {
  return {a.re * b.re - a.im * b.im, a.re * b.im + a.im * b.re};
}
__device__ __forceinline__ C2 cadd(C2 a, C2 b) { return {a.re + b.re, a.im + b.im}; }

__device__ __forceinline__ float fast_tanh(float x) {
#if __has_builtin(__builtin_amdgcn_tanhf)
  return __builtin_amdgcn_tanhf(x);           // v_tanh_f32 (CDNA5 trans op)
#elif __has_builtin(__builtin_amdgcn_tanh_f32)
  return __builtin_amdgcn_tanh_f32(x);
#else
  // tanh(x) = 1 - 2/(exp(2x)+1), via hardware exp2 + rcp (saturates correctly)
  float e = __builtin_amdgcn_exp2f(2.88539008177793f * x);  // 2*log2(e)
  return 1.0f - 2.0f * __builtin_amdgcn_rcpf(e + 1.0f);
#endif
}

// ---------------------------------------------------------------------------
// Setup kernel: one wave. Builds U (4x4 complex) from weights, then
// Q = [M_re^T | M_im^T](4x8) x [D M_re ; D M_im](8x4) on the WMMA unit,
// then reduces Q to the 3x3 matrix S' (absorbing action = (1+z0)/2).
// ---------------------------------------------------------------------------
__global__ __launch_bounds__(32) void qpol_setup(const float* __restrict__ w,
                                                 float* __restrict__ Sout) {
  __shared__ float Pl[4][8];    // A operand source (4x8, rows padded to 16)
  __shared__ float Wl[8][4];    // B operand source (8x4, cols padded to 16)
  __shared__ float Ql[4][16];   // Q extraction (cols 0..3 meaningful)

  const int lane = threadIdx.x;

  if (lane == 0) {
    // ---- build U = (layer gates + CNOTs) applied to identity ----
    C2 U[4][4];
    for (int r = 0; r < 4; ++r)
      for (int c = 0; c < 4; ++c) U[r][c] = {(r == c) ? 1.0f : 0.0f, 0.0f};

    for (int l = 0; l < N_LAYERS; ++l) {
      for (int q = 0; q < 2; ++q) {
        float ty = w[(l * 2 + q) * 2 + 0] * 0.5f;   // RY(theta)/2
        float tz = w[(l * 2 + q) * 2 + 1] * 0.5f;   // RZ(theta)/2
        float cy = cosf(ty), sy = sinf(ty);
        float cz = cosf(tz), sz = sinf(tz);
        // g = RZ @ RY ; em = e^{-i tz}, ep = e^{+i tz}
        C2 g00 = { cy * cz, -cy * sz};
        C2 g01 = {-sy * cz,  sy * sz};
        C2 g10 = { sy * cz,  sy * sz};
        C2 g11 = { cy * cz,  cy * sz};
        C2 V[4][4];
        if (q == 0) {  // act on qubit 0 (row index bit 1)
          for (int c = 0; c < 4; ++c)
            for (int i1 = 0; i1 < 2; ++i1) {
              C2 a = U[0 * 2 + i1][c], b = U[1 * 2 + i1][c];
              V[0 * 2 + i1][c] = cadd(cmul(g00, a), cmul(g01, b));
              V[1 * 2 + i1][c] = cadd(cmul(g10, a), cmul(g11, b));
            }
        } else {       // act on qubit 1 (row index bit 0)
          for (int c = 0; c < 4; ++c)
            for (int i0 = 0; i0 < 2; ++i0) {
              C2 a = U[i0 * 2 + 0][c], b = U[i0 * 2 + 1][c];
              V[i0 * 2 + 0][c] = cadd(cmul(g00, a), cmul(g01, b));
              V[i0 * 2 + 1][c] = cadd(cmul(g10, a), cmul(g11, b));
            }
        }
        for (int r = 0; r < 4; ++r)
          for (int c = 0; c < 4; ++c) U[r][c] = V[r][c];
      }
      // CNOT(ctrl=0,tgt=1): swap rows 2,3 ; CNOT(ctrl=1,tgt=0): swap rows 1,3
      for (int c = 0; c < 4; ++c) { C2 t = U[2][c]; U[2][c] = U[3][c]; U[3][c] = t; }
      for (int c = 0; c < 4; ++c) { C2 t = U[1][c]; U[1][c] = U[3][c]; U[3][c] = t; }
    }

    // out[r] = sum_k A[r][k]*x_k  with A[.,0]=U0, A[.,1]=-i*U1, A[.,2]=-i*U2, A[.,3]=-U3
    float Mre[4][4], Mim[4][4];
    for (int r = 0; r < 4; ++r) {
      Mre[r][0] =  U[r][0].re;  Mim[r][0] =  U[r][0].im;
      Mre[r][1] =  U[r][1].im;  Mim[r][1] = -U[r][1].re;
      Mre[r][2] =  U[r][2].im;  Mim[r][2] = -U[r][2].re;
      Mre[r][3] = -U[r][3].re;  Mim[r][3] = -U[r][3].im;
    }
    const float D[4] = {1.0f, 1.0f, -1.0f, -1.0f};   // <Z0> signs per row
    for (int a = 0; a < 4; ++a)
      for (int j = 0; j < 8; ++j)
        Pl[a][j] = (j < 4) ? Mre[j][a] : Mim[j - 4][a];
    for (int j = 0; j < 8; ++j)
      for (int b = 0; b < 4; ++b)
        Wl[j][b] = (j < 4) ? D[j] * Mre[j][b] : D[j - 4] * Mim[j - 4][b];
  }
  __syncthreads();

#if __has_builtin(__builtin_amdgcn_wmma_f32_16x16x4_f32)
  // Q(4x4) = P(4x8) x W(8x4), padded to 16x16, K accumulated in 2 chunks of 4.
  // A 16x4 f32 layout: lanes 0-15 row M, v0:K=0 v1:K=1; lanes 16-31 v0:K=2 v1:K=3.
  // B 4x16 f32 layout assumed mirror: v0 holds K=0 (lanes 0-15) / K=2 (16-31),
  // v1 holds K=1 / K=3, N = lane%16.
  v8f acc = {};
  const int half = lane >> 4;
  const int rc   = lane & 15;           // A row (M) and B col (N)
  for (int kk = 0; kk < 2; ++kk) {
    const int k0 = kk * 4 + half * 2;
    v2f a, b;
    a.x = (rc < 4) ? Pl[rc][k0 + 0] : 0.0f;
    a.y = (rc < 4) ? Pl[rc][k0 + 1] : 0.0f;
    b.x = (rc < 4) ? Wl[k0 + 0][rc] : 0.0f;
    b.y = (rc < 4) ? Wl[k0 + 1][rc] : 0.0f;
    acc = __builtin_amdgcn_wmma_f32_16x16x4_f32(false, a, false, b,
                                                (short)0, acc, false, false);
  }
  // C layout: VGPR r <-> M=r (lanes 0-15, N=lane). Rows 0..3 carry Q.
  if (lane < 16) {
    Ql[0][lane] = acc.s0;
    Ql[1][lane] = acc.s1;
    Ql[2][lane] = acc.s2;
    Ql[3][lane] = acc.s3;
  }
#else
  if (lane == 0) {
    for (int a = 0; a < 4; ++a)
      for (int b = 0; b < 4; ++b) {
        float s = 0.0f;
        for (int j = 0; j < 8; ++j) s += Pl[a][j] * Wl[j][b];
        Ql[a][b] = s;
      }
  }
#endif
  __syncthreads();

  if (lane == 0) {
    // Double-angle bilinear reduction: u_i*u_k = au0 + au1*cos(t0) + au2*sin(t0)
    // (0,0)->(1/2,1/2,0)  (0,1),(1,0)->(0,0,1/2)  (1,1)->(1/2,-1/2,0)
    float S3[3][3] = {};
    for (int i = 0; i < 2; ++i)
      for (int k = 0; k < 2; ++k)
        for (int j = 0; j < 2; ++j)
          for (int m = 0; m < 2; ++m) {
            float q = Ql[2 * i + j][2 * k + m];
            float au[3] = {(i == k) ? 0.5f : 0.0f,
                           (i == k) ? (i ? -0.5f : 0.5f) : 0.0f,
                           (i == k) ? 0.0f : 0.5f};
            float av[3] = {(j == m) ? 0.5f : 0.0f,
                           (j == m) ? (j ? -0.5f : 0.5f) : 0.0f,
                           (j == m) ? 0.0f : 0.5f};
            for (int a = 0; a < 3; ++a)
              for (int b = 0; b < 3; ++b) S3[a][b] += q * au[a] * av[b];
          }
    // action = (1 + z0)/2  -> fold into S'
    for (int a = 0; a < 3; ++a)
      for (int b = 0; b < 3; ++b) {
        float v = 0.5f * S3[a][b];
        if (a == 0 && b == 0) v += 0.5f;
        Sout[a * 3 + b] = v;
      }
  }
}

// ---------------------------------------------------------------------------
// Streaming kernel: memory-bound elementwise pass.
// ---------------------------------------------------------------------------
__global__ __launch_bounds__(256) void qpol_main(const float4* __restrict__ st,
                                                 const float* __restrict__ S,
                                                 float* __restrict__ out, int n) {
  // 9 uniform coefficients, hoisted out of the loop.
  const float s00 = S[0], s01 = S[1], s02 = S[2];
  const float s10 = S[3], s11 = S[4], s12 = S[5];
  const float s20 = S[6], s21 = S[7], s22 = S[8];

  const int stride = gridDim.x * blockDim.x;
  for (int i = blockIdx.x * blockDim.x + threadIdx.x; i < n; i += stride) {
    __builtin_prefetch(&st[i + stride], 0, 1);   // global_prefetch_b8 (speculative)
    float4 r = st[i];                             // global_load_b128 (row of state)
    float t0 = fast_tanh(r.x);
    float t1 = fast_tanh(r.y);
    // v_sin/cos_f32 take revolutions: theta = pi*t -> rev = t/2
    float C0 = __builtin_amdgcn_cosf(0.5f * t0);
    float S0 = __builtin_amdgcn_sinf(0.5f * t0);
    float C1 = __builtin_amdgcn_cosf(0.5f * t1);
    float S1 = __builtin_amdgcn_sinf(0.5f * t1);
    float g0 = s00 + s01 * C1 + s02 * S1;
    float g1 = s10 + s11 * C1 + s12 * S1;
    float g2 = s20 + s21 * C1 + s22 * S1;
    out[i] = g0 + g1 * C0 + g2 * S0;
  }
}

extern "C" void kernel_launch(void* const* d_in, const int* in_sizes, int n_in,
                              void* d_out, int out_size, void* d_ws, size_t ws_size,
                              hipStream_t stream) {
  (void)n_in; (void)ws_size; (void)out_size;
  const float*  state   = (const float*)d_in[0];   // [B,4] f32
  const float*  weights = (const float*)d_in[1];   // [4,2,2] f32
  float*        out     = (float*)d_out;           // [B,1] f32
  float*        Smat    = (float*)d_ws;            // 9 floats scratch

  const int n = in_sizes[0] / 4;                   // batch rows

  qpol_setup<<<1, 32, 0, stream>>>(weights, Smat);

  const int block = 256;
  int blocks = (n + block * 8 - 1) / (block * 8);  // ~8 elems/thread
  if (blocks < 1) blocks = 1;
  qpol_main<<<blocks, block, 0, stream>>>((const float4*)state, Smat, out, n);
}